// Mixtral_72851235275310
// MI455X (gfx1250) — compile-verified
//
#include <hip/hip_runtime.h>
#include <hip/hip_bf16.h>
#include <math.h>

// MI455X (gfx1250) fp32-WMMA transformer forward.
// - All GEMMs on v_wmma_f32_16x16x4_f32 (exact fp32 vs reference; with every
//   weight read exactly once from HBM (~700MB fp32), the kernel is memory
//   bound (~30us floor @ 23.3 TB/s) and matrix-f32 throughput is ample).
// - Weight tiles (the dominant traffic) are streamed into LDS by the Tensor
//   Data Mover (tensor_load_to_lds, TENSORcnt double-buffered); the D#
//   pad_enable feature reproduces the bank-conflict-free stride-17 LDS
//   layout in hardware. Vector-load fallback for boundary tiles/toolchains.

typedef float v2f __attribute__((ext_vector_type(2)));
typedef float v8f __attribute__((ext_vector_type(8)));
typedef unsigned int u32x4 __attribute__((ext_vector_type(4)));
typedef int i32x4 __attribute__((ext_vector_type(4)));
typedef int i32x8 __attribute__((ext_vector_type(8)));

#define T_SEQ   1024
#define DMODEL  1024
#define NHEAD   16
#define HDIM    64
#define NEXP    8
#define FFDIM   4096
#define VOUT_N  31999
#define QKV_LD  3072
#define WIN     512
#define NEG_INF (-__builtin_inff())

#if defined(__AMDGCN__) && defined(__has_builtin)
#  if __has_builtin(__builtin_amdgcn_tensor_load_to_lds)
#    define HAVE_TDM 1
#  endif
#endif
#ifndef HAVE_TDM
#  define HAVE_TDM 0
#endif

static __device__ __forceinline__ float wave_reduce_add(float v) {
#pragma unroll
  for (int off = 16; off >= 1; off >>= 1) v += __shfl_xor(v, off, 32);
  return v;
}

#if HAVE_TDM
// Issue a TDM DMA of a 64(row) x 16(col) fp32 tile, row stride ldw elements,
// into LDS at lds_addr with 1 DWORD of padding after every 16 DWORDs
// (row pitch 17 floats == sW layout). One instruction per wave; TENSORcnt.
static __device__ __forceinline__ void tdm_load_tile(const float* gsrc,
                                                     int ldw,
                                                     unsigned lds_addr) {
  unsigned long long ga = (unsigned long long)(uintptr_t)gsrc;
  u32x4 g0;
  g0[0] = 1u;                               // count=1, no gather/iterate
  g0[1] = lds_addr;                         // D# lds_addr
  g0[2] = (unsigned)ga;                     // global_addr[31:0]
  g0[3] = (unsigned)(ga >> 32) | (2u << 30);// global_addr[56:32] | type=2
  i32x8 g1;
  g1[0] = (2 << 16) | (1 << 20) | (3 << 22);// data_size=4B, pad_en, 16 dw/pad
  g1[1] = (int)(16u << 16);                 // tensor_dim0 = 16 (tile-exact)
  g1[2] = (int)(64u << 16);                 // tensor_dim1 = 64
  g1[3] = (int)(16u << 16);                 // tile_dim0 = 16
  g1[4] = 64;                               // tile_dim1 = 64
  g1[5] = ldw;                              // tensor_dim0_stride (elements)
  g1[6] = 0;
  g1[7] = 0;
  i32x4 z4 = {0, 0, 0, 0};
#if __clang_major__ >= 23
  i32x8 z8 = {0, 0, 0, 0, 0, 0, 0, 0};
  __builtin_amdgcn_tensor_load_to_lds(g0, g1, z4, z4, z8, 0);
#else
  __builtin_amdgcn_tensor_load_to_lds(g0, g1, z4, z4, 0);
#endif
}
#endif

// ---------------------------------------------------------------------------
// Embedding gather: x[t][:] = emb[idx[t]][:]
// ---------------------------------------------------------------------------
__global__ __launch_bounds__(256) void embed_kernel(
    const int* __restrict__ idx, const float* __restrict__ emb,
    float* __restrict__ x) {
  int t = blockIdx.x, tid = threadIdx.x;
  long row = idx[t];
  reinterpret_cast<float4*>(x + (long)t * DMODEL)[tid] =
      reinterpret_cast<const float4*>(emb + row * DMODEL)[tid];
}

// ---------------------------------------------------------------------------
// RMSNorm (fp32, eps=1e-6): out = x * rsqrt(mean(x^2)+eps) * g
// ---------------------------------------------------------------------------
__global__ __launch_bounds__(256) void rmsnorm_kernel(
    const float* __restrict__ x, const float* __restrict__ g,
    float* __restrict__ out) {
  int t = blockIdx.x, tid = threadIdx.x;
  int lane = tid & 31, wave = tid >> 5;
  float4 v = reinterpret_cast<const float4*>(x + (long)t * DMODEL)[tid];
  float ss = v.x * v.x + v.y * v.y + v.z * v.z + v.w * v.w;
  ss = wave_reduce_add(ss);
  __shared__ float red[8];
  if (lane == 0) red[wave] = ss;
  __syncthreads();
  if (tid == 0) {
    float s2 = 0.f;
#pragma unroll
    for (int i = 0; i < 8; ++i) s2 += red[i];
    red[0] = rsqrtf(s2 * (1.0f / DMODEL) + 1e-6f);
  }
  __syncthreads();
  float rinv = red[0];
  float4 gg = reinterpret_cast<const float4*>(g)[tid];
  float4 ov;
  ov.x = v.x * rinv * gg.x; ov.y = v.y * rinv * gg.y;
  ov.z = v.z * rinv * gg.z; ov.w = v.w * rinv * gg.w;
  reinterpret_cast<float4*>(out + (long)t * DMODEL)[tid] = ov;
}

// ---------------------------------------------------------------------------
// RoPE on normalized input (reference ropes BEFORE the projections; q == k).
// ---------------------------------------------------------------------------
__global__ __launch_bounds__(256) void rope_kernel(
    const float* __restrict__ xn, float* __restrict__ r) {
  int t = blockIdx.x, tid = threadIdx.x;
  const float* xr = xn + (long)t * DMODEL;
  float* rr = r + (long)t * DMODEL;
#pragma unroll
  for (int it = 0; it < 2; ++it) {
    int pp = tid + it * 256;            // pair index 0..511
    int hh = pp >> 5;                   // head (32 pairs per head)
    int p = pp & 31;                    // pair within head
    int base = hh * HDIM + 2 * p;
    float x1 = xr[base], x2 = xr[base + 1];
    float theta = expf(-(float)(2 * p) * (9.2103403719761836f / 64.0f));
    float ang = (float)t * theta;
    float s = sinf(ang), c = cosf(ang);
    rr[base]     = x1 * c - x2 * s;
    rr[base + 1] = x1 * s + x2 * c;
  }
}

// ---------------------------------------------------------------------------
// Tiled WMMA GEMM:  C[m][n] = epilogue( sum_k A[m][k] * W[n][k] + bias[n] )
// Block tile 32x64, 8 waves (2x4), each wave one 16x16 v_wmma_f32_16x16x4_f32
// accumulator. Weight tiles stream via TDM (double buffered, TENSORcnt);
// activation tiles via vector loads. A rows optionally gathered; M optionally
// from a device counter (per-expert token counts).
// mode 0: plain   mode 1: SiLU   mode 2: gated scatter to ybuf[(2t+s)][n]
// ---------------------------------------------------------------------------
__global__ __launch_bounds__(256) void gemm_wmma_kernel(
    const float* __restrict__ A, const float* __restrict__ W,
    const float* __restrict__ bias, float* __restrict__ C,
    int M, int N, int K, int lda, int ldw, int ldc, int mode,
    const int* __restrict__ gather, const int* __restrict__ cntPtr,
    const int* __restrict__ tl, const int* __restrict__ sl,
    const float* __restrict__ gl) {
  const int BM = 32, BN = 64, BK = 16;
  __shared__ float sA[BM][BK + 1];
  __shared__ float sW[2][BN][BK + 1];
  int Meff = cntPtr ? cntPtr[0] : M;
  int bm = blockIdx.x * BM;
  if (bm >= Meff) return;                 // uniform per block (EXEC stays full)
  int bn = blockIdx.y * BN;
  int tid = threadIdx.x, lane = tid & 31, wave = tid >> 5;
  int wr = wave >> 2, wc = wave & 3;      // 2 x 4 wave grid
  int nn = lane & 15, half = lane >> 4, kk0 = 2 * half;
  v8f acc = {};
  int nsteps = K / BK;                    // K is always a multiple of 16 here

#if HAVE_TDM
  bool tdm = (bn + BN <= N);              // boundary tiles take vector path
  unsigned ldsW[2];
  ldsW[0] = (unsigned)(uintptr_t)(&sW[0][0][0]);
  ldsW[1] = (unsigned)(uintptr_t)(&sW[1][0][0]);
  if (tdm && wave == 0)                   // prologue DMA: tile 0 -> buf 0
    tdm_load_tile(W + (unsigned)bn * ldw, ldw, ldsW[0]);
#else
  const bool tdm = false;
#endif

  for (int s = 0; s < nsteps; ++s) {
    int k0 = s * BK;
    int buf = s & 1;
    // activation tile (optionally gathered rows)
    for (int i = tid; i < BM * BK; i += 256) {
      int r = i >> 4, c = i & 15, gr = bm + r;
      float v = 0.f;
      if (gr < Meff) {
        int ar = gather ? gather[gr] : gr;
        v = A[(unsigned)(ar * lda + k0 + c)];
      }
      sA[r][c] = v;
    }
    if (tdm) {
#if HAVE_TDM
      if (wave == 0) {                    // DMA next tile, wait for current
        if (s + 1 < nsteps) {
          tdm_load_tile(W + (unsigned)bn * ldw + (unsigned)(k0 + BK), ldw,
                        ldsW[buf ^ 1]);
          __builtin_amdgcn_s_wait_tensorcnt(1);
        } else {
          __builtin_amdgcn_s_wait_tensorcnt(0);
        }
      }
#endif
    } else {
      for (int i = tid; i < BN * BK; i += 256) {
        int r = i >> 4, c = i & 15, gn = bn + r;
        sW[buf][r][c] = (gn < N) ? W[(unsigned)(gn * ldw + k0 + c)] : 0.f;
      }
      if (k0 + BK < K && tid < BN) {
        int gn = bn + tid;
        if (gn < N) __builtin_prefetch(&W[(unsigned)(gn * ldw + k0 + BK)], 0, 0);
      }
    }
    __syncthreads();
#pragma unroll
    for (int kc = 0; kc < 4; ++kc) {
      v2f a, b;
      a.x = sA[wr * 16 + nn][kc * 4 + kk0];
      a.y = sA[wr * 16 + nn][kc * 4 + kk0 + 1];
      b.x = sW[buf][wc * 16 + nn][kc * 4 + kk0];
      b.y = sW[buf][wc * 16 + nn][kc * 4 + kk0 + 1];
      acc = __builtin_amdgcn_wmma_f32_16x16x4_f32(false, a, false, b,
                                                  (short)0, acc, false, false);
    }
    __syncthreads();
  }
  int col = bn + wc * 16 + nn;
  if (col < N) {
    float bv = bias ? bias[col] : 0.f;
#pragma unroll
    for (int r = 0; r < 8; ++r) {
      int row = bm + wr * 16 + r + 8 * half;
      if (row < Meff) {
        float v = acc[r] + bv;
        if (mode == 1) v = v / (1.0f + __expf(-v));          // SiLU
        if (mode == 2) {
          int t = tl[row], s2 = sl[row];
          C[(unsigned)((t * 2 + s2) * ldc + col)] = gl[row] * v;
        } else {
          C[(long)row * ldc + col] = v;
        }
      }
    }
  }
}

// ---------------------------------------------------------------------------
// Fused sliding-window flash attention, fp32 WMMA.
// 1 wave per (16-query tile, head). qkv layout: [T][3*D], q|k|v column bands.
// ---------------------------------------------------------------------------
__global__ __launch_bounds__(32) void attn_kernel(
    const float* __restrict__ qkv, float* __restrict__ o) {
  int qt = blockIdx.x, h = blockIdx.y;
  int q0 = qt * 16;
  int lane = threadIdx.x;
  int nn = lane & 15, half = lane >> 4, kk0 = 2 * half;
  __shared__ float sP[16][17];

  // Q fragments for the whole tile (A operand, 16 k-chunks of 4)
  v2f qf[16];
  {
    const float* qrow = qkv + (long)(q0 + nn) * QKV_LD + h * HDIM;
#pragma unroll
    for (int kc = 0; kc < 16; ++kc) {
      float2 t2 = *reinterpret_cast<const float2*>(qrow + kc * 4 + kk0);
      qf[kc].x = t2.x; qf[kc].y = t2.y;
    }
  }
  v8f zero = {};
  v8f of[4] = {zero, zero, zero, zero};
  float m_r[8], l_r[8];
#pragma unroll
  for (int r = 0; r < 8; ++r) { m_r[r] = NEG_INF; l_r[r] = 0.f; }

  int ktlo = q0 - (WIN - 1);
  if (ktlo < 0) ktlo = 0;
  ktlo >>= 4;
  for (int kt = ktlo; kt <= qt; ++kt) {
    int k0 = kt * 16;
    // S = Q K^T  (16 chained f32 WMMAs over HD=64)
    v8f s = zero;
    {
      const float* krow = qkv + (long)(k0 + nn) * QKV_LD + DMODEL + h * HDIM;
#pragma unroll
      for (int kc = 0; kc < 16; ++kc) {
        float2 t2 = *reinterpret_cast<const float2*>(krow + kc * 4 + kk0);
        v2f b; b.x = t2.x; b.y = t2.y;
        s = __builtin_amdgcn_wmma_f32_16x16x4_f32(false, qf[kc], false, b,
                                                  (short)0, s, false, false);
      }
    }
    // online softmax with sliding-window mask, in C-fragment layout
#pragma unroll
    for (int r = 0; r < 8; ++r) {
      int i = q0 + r + 8 * half;       // query index of this element
      int j = k0 + nn;                 // key index
      float sv = s[r] * 0.125f;        // 1/sqrt(64)
      if (j > i || (j + WIN) <= i) sv = NEG_INF;
      float tm = sv;
#pragma unroll
      for (int off = 8; off >= 1; off >>= 1)
        tm = fmaxf(tm, __shfl_xor(tm, off, 16));
      float mnew = fmaxf(m_r[r], tm);
      float p = (sv == NEG_INF) ? 0.f : __expf(sv - mnew);
      float alpha = (m_r[r] == NEG_INF) ? 0.f : __expf(m_r[r] - mnew);
      float ps = p;
#pragma unroll
      for (int off = 8; off >= 1; off >>= 1) ps += __shfl_xor(ps, off, 16);
      l_r[r] = l_r[r] * alpha + ps;
      m_r[r] = mnew;
#pragma unroll
      for (int nt = 0; nt < 4; ++nt) of[nt][r] *= alpha;
      sP[r + 8 * half][nn] = p;        // stage P for re-entry as A operand
    }
    __syncthreads();
    // O += P V  (A = P from LDS, B = V tiles)
    {
      const float* vbase = qkv + 2 * DMODEL + h * HDIM;
#pragma unroll
      for (int kc = 0; kc < 4; ++kc) {
        v2f a;
        a.x = sP[nn][kc * 4 + kk0];
        a.y = sP[nn][kc * 4 + kk0 + 1];
#pragma unroll
        for (int nt = 0; nt < 4; ++nt) {
          v2f b;
          b.x = vbase[(long)(k0 + kc * 4 + kk0) * QKV_LD + nt * 16 + nn];
          b.y = vbase[(long)(k0 + kc * 4 + kk0 + 1) * QKV_LD + nt * 16 + nn];
          of[nt] = __builtin_amdgcn_wmma_f32_16x16x4_f32(false, a, false, b,
                                                         (short)0, of[nt],
                                                         false, false);
        }
      }
    }
    __syncthreads();
  }
  // normalize and store [T][D]
  float* orow = o + h * HDIM;
#pragma unroll
  for (int r = 0; r < 8; ++r) {
    float inv = 1.0f / l_r[r];
    int row = q0 + r + 8 * half;
#pragma unroll
    for (int nt = 0; nt < 4; ++nt)
      orow[(long)row * DMODEL + nt * 16 + nn] = of[nt][r] * inv;
  }
}

// ---------------------------------------------------------------------------
// Router: logits = xn2 @ rw^T + rb ; top-2 (jax tie-break: lower index) ;
// gates = softmax of the two logits.
// ---------------------------------------------------------------------------
__global__ __launch_bounds__(256) void router_kernel(
    const float* __restrict__ xn2, const float* __restrict__ rw,
    const float* __restrict__ rb, int* __restrict__ topi,
    float* __restrict__ topg) {
  int t = blockIdx.x, tid = threadIdx.x, lane = tid & 31, e = tid >> 5;
  const float* xr = xn2 + (long)t * DMODEL;
  const float* we = rw + (long)e * DMODEL;
  float s = 0.f;
  for (int k = lane; k < DMODEL; k += 32) s += xr[k] * we[k];
  s = wave_reduce_add(s);
  __shared__ float slog[8];
  if (lane == 0) slog[e] = s + rb[e];
  __syncthreads();
  if (tid == 0) {
    float v0 = NEG_INF; int i0 = 0;
    for (int q = 0; q < 8; ++q) if (slog[q] > v0) { v0 = slog[q]; i0 = q; }
    float v1 = NEG_INF; int i1 = 0;
    for (int q = 0; q < 8; ++q)
      if (q != i0 && slog[q] > v1) { v1 = slog[q]; i1 = q; }
    float e1 = __expf(v1 - v0);
    float den = 1.0f + e1;
    topi[2 * t] = i0; topi[2 * t + 1] = i1;
    topg[2 * t] = 1.0f / den; topg[2 * t + 1] = e1 / den;
  }
}

// Deterministic per-expert token lists (sequential scan per expert).
__global__ void build_lists_kernel(
    const int* __restrict__ topi, const float* __restrict__ topg,
    int* __restrict__ tl, int* __restrict__ sl, float* __restrict__ gl,
    int* __restrict__ cnt) {
  int e = threadIdx.x;
  if (e >= NEXP) return;
  int c = 0;
  for (int t = 0; t < T_SEQ; ++t) {
    int i0 = topi[2 * t], i1 = topi[2 * t + 1];
    if (i0 == e) {
      tl[e * T_SEQ + c] = t; sl[e * T_SEQ + c] = 0;
      gl[e * T_SEQ + c] = topg[2 * t]; ++c;
    } else if (i1 == e) {
      tl[e * T_SEQ + c] = t; sl[e * T_SEQ + c] = 1;
      gl[e * T_SEQ + c] = topg[2 * t + 1]; ++c;
    }
  }
  cnt[e] = c;
}

// x[t] += ybuf[t][0] + ybuf[t][1]  (fixed order -> deterministic)
__global__ __launch_bounds__(256) void combine_kernel(
    float* __restrict__ x, const float* __restrict__ ybuf) {
  int t = blockIdx.x, tid = threadIdx.x;
  float4* xp = reinterpret_cast<float4*>(x + (long)t * DMODEL);
  const float4* y0 = reinterpret_cast<const float4*>(ybuf + (long)(2 * t) * DMODEL);
  const float4* y1 = reinterpret_cast<const float4*>(ybuf + (long)(2 * t + 1) * DMODEL);
  float4 a = xp[tid], b0 = y0[tid], b1 = y1[tid];
  a.x += b0.x + b1.x; a.y += b0.y + b1.y;
  a.z += b0.z + b1.z; a.w += b0.w + b1.w;
  xp[tid] = a;
}

// ---------------------------------------------------------------------------
extern "C" void kernel_launch(void* const* d_in, const int* in_sizes, int n_in,
                              void* d_out, int out_size, void* d_ws,
                              size_t ws_size, hipStream_t stream) {
  (void)in_sizes; (void)n_in; (void)out_size; (void)ws_size;
  const int*   idx  = (const int*)d_in[0];
  const float* emb  = (const float*)d_in[1];
  const float* rmsS = (const float*)d_in[2];
  const float* frms = (const float*)d_in[3];
  const float* inw  = (const float*)d_in[4];
  const float* inb  = (const float*)d_in[5];
  const float* outw = (const float*)d_in[6];
  const float* outb = (const float*)d_in[7];
  const float* rw   = (const float*)d_in[8];
  const float* rb   = (const float*)d_in[9];
  const float* w1   = (const float*)d_in[10];
  const float* b1   = (const float*)d_in[11];
  const float* w2   = (const float*)d_in[12];
  const float* b2   = (const float*)d_in[13];
  const float* lmw  = (const float*)d_in[14];
  const float* lmb  = (const float*)d_in[15];
  float* logits = (float*)d_out;

  char* ws = (char*)d_ws;
  size_t off = 0;
  auto alloc = [&](size_t bytes) -> void* {
    void* p = (void*)(ws + off);
    off += (bytes + 255) & ~(size_t)255;
    return p;
  };
  float* x    = (float*)alloc((size_t)T_SEQ * DMODEL * 4);
  float* xn   = (float*)alloc((size_t)T_SEQ * DMODEL * 4);
  float* xn2  = (float*)alloc((size_t)T_SEQ * DMODEL * 4);
  float* rbuf = (float*)alloc((size_t)T_SEQ * DMODEL * 4);
  float* qkv  = (float*)alloc((size_t)T_SEQ * QKV_LD * 4);
  float* obuf = (float*)alloc((size_t)T_SEQ * DMODEL * 4);
  float* hbuf = (float*)alloc((size_t)T_SEQ * FFDIM * 4);
  float* ybuf = (float*)alloc((size_t)T_SEQ * 2 * DMODEL * 4);
  int*   tl   = (int*)alloc((size_t)NEXP * T_SEQ * 4);
  int*   slb  = (int*)alloc((size_t)NEXP * T_SEQ * 4);
  float* gl   = (float*)alloc((size_t)NEXP * T_SEQ * 4);
  int*   cnt  = (int*)alloc(NEXP * 4);
  int*   topi = (int*)alloc((size_t)2 * T_SEQ * 4);
  float* topg = (float*)alloc((size_t)2 * T_SEQ * 4);

  auto gemm = [&](const float* A, const float* W, const float* bias, float* C,
                  int M, int N, int K, int lda, int ldw, int ldc, int mode,
                  const int* gather, const int* cntPtr, const int* tlp,
                  const int* slp, const float* glp) {
    dim3 grid((M + 31) / 32, (N + 63) / 64);
    hipLaunchKernelGGL(gemm_wmma_kernel, grid, dim3(256), 0, stream, A, W,
                       bias, C, M, N, K, lda, ldw, ldc, mode, gather, cntPtr,
                       tlp, slp, glp);
  };

  embed_kernel<<<T_SEQ, 256, 0, stream>>>(idx, emb, x);

  for (int blk = 0; blk < 2; ++blk) {
    const float* g = rmsS + blk * DMODEL;
    rmsnorm_kernel<<<T_SEQ, 256, 0, stream>>>(x, g, xn);
    rope_kernel<<<T_SEQ, 256, 0, stream>>>(xn, rbuf);
    // q/k projections (identical roped input r), then v projection (input xn)
    gemm(rbuf, inw + (long)blk * 3 * DMODEL * DMODEL, inb + blk * 3 * DMODEL,
         qkv, T_SEQ, 2 * DMODEL, DMODEL, DMODEL, DMODEL, QKV_LD, 0,
         nullptr, nullptr, nullptr, nullptr, nullptr);
    gemm(xn, inw + (long)blk * 3 * DMODEL * DMODEL + (long)2 * DMODEL * DMODEL,
         inb + blk * 3 * DMODEL + 2 * DMODEL, qkv + 2 * DMODEL,
         T_SEQ, DMODEL, DMODEL, DMODEL, DMODEL, QKV_LD, 0,
         nullptr, nullptr, nullptr, nullptr, nullptr);
    attn_kernel<<<dim3(T_SEQ / 16, NHEAD), 32, 0, stream>>>(qkv, obuf);
    gemm(obuf, outw + (long)blk * DMODEL * DMODEL, outb + blk * DMODEL, x,
         T_SEQ, DMODEL, DMODEL, DMODEL, DMODEL, DMODEL, 0,
         nullptr, nullptr, nullptr, nullptr, nullptr);   // no attn residual
    // MoE
    rmsnorm_kernel<<<T_SEQ, 256, 0, stream>>>(x, g, xn2);
    router_kernel<<<T_SEQ, 256, 0, stream>>>(xn2, rw + blk * NEXP * DMODEL,
                                             rb + blk * NEXP, topi, topg);
    build_lists_kernel<<<1, 8, 0, stream>>>(topi, topg, tl, slb, gl, cnt);
    for (int e = 0; e < NEXP; ++e) {
      long we = (long)(blk * NEXP + e);
      gemm(xn2, w1 + we * FFDIM * DMODEL, b1 + we * FFDIM, hbuf,
           T_SEQ, FFDIM, DMODEL, DMODEL, DMODEL, FFDIM, 1,
           tl + e * T_SEQ, cnt + e, nullptr, nullptr, nullptr);
      gemm(hbuf, w2 + we * DMODEL * FFDIM, b2 + we * DMODEL, ybuf,
           T_SEQ, DMODEL, FFDIM, FFDIM, FFDIM, DMODEL, 2,
           nullptr, cnt + e, tl + e * T_SEQ, slb + e * T_SEQ, gl + e * T_SEQ);
    }
    combine_kernel<<<T_SEQ, 256, 0, stream>>>(x, ybuf);
  }

  rmsnorm_kernel<<<T_SEQ, 256, 0, stream>>>(x, frms, xn);
  gemm(xn, lmw, lmb, logits, T_SEQ, VOUT_N, DMODEL, DMODEL, DMODEL, VOUT_N, 0,
       nullptr, nullptr, nullptr, nullptr, nullptr);
}